// Attentioncross_23227183136847
// MI455X (gfx1250) — compile-verified
//
#include <hip/hip_runtime.h>

// Fused ring-neighbor attention for MI455X (gfx1250, wave32).
// out[n,r,:] = z[n,r,:] + b1*z[n,k1,:] + b2*z[n,k2,:]
// betas from softmax over scores of rows (r,k1,k2);
// score(row) = W2 . leaky_relu(W1 . z[row]) via V_WMMA_F32_16X16X4_F32.
// Memory-bound: z read once (async DMA to LDS), out written once.

typedef float v2f __attribute__((ext_vector_type(2)));
typedef float v8f __attribute__((ext_vector_type(8)));

#define DIM           256
#define RNUM          8
#define STRIDE        260   // padded LDS row stride: 260%64==4 -> conflict-free WMMA-A fetch
#define WAVES         8
#define ROWS_PER_WAVE 16    // 2 samples per wave (rows of one sample contiguous)
#define ZT_FLOATS     (ROWS_PER_WAVE * STRIDE)       // 4160
#define PW_FLOATS     (ZT_FLOATS + 16 + 32)          // tile + scores + betas = 4208

extern __shared__ float lds[];

// Wave-local LDS fence: all sharing is intra-wave, so no s_barrier needed.
__device__ __forceinline__ void wave_fence_ds() {
    __builtin_amdgcn_wave_barrier();
    asm volatile("s_wait_dscnt 0x0" ::: "memory");
    __builtin_amdgcn_wave_barrier();
}

// Wait for async global->LDS DMA (ASYNCcnt) before touching the tile.
__device__ __forceinline__ void wave_fence_async() {
    __builtin_amdgcn_wave_barrier();
    asm volatile("s_wait_asynccnt 0x0" ::: "memory");
    __builtin_amdgcn_wave_barrier();
}

// Butterfly sum within each 16-lane half (wave32) via ds_swizzle XOR patterns.
__device__ __forceinline__ float half16_reduce_add(float t) {
    t += __int_as_float(__builtin_amdgcn_ds_swizzle(__float_as_int(t), 0x041f)); // xor 1
    t += __int_as_float(__builtin_amdgcn_ds_swizzle(__float_as_int(t), 0x081f)); // xor 2
    t += __int_as_float(__builtin_amdgcn_ds_swizzle(__float_as_int(t), 0x101f)); // xor 4
    t += __int_as_float(__builtin_amdgcn_ds_swizzle(__float_as_int(t), 0x201f)); // xor 8
    return t;
}

__global__ __launch_bounds__(WAVES * 32, 1)
void ring_attn_fused(const float* __restrict__ z,
                     const float* __restrict__ W1,
                     const float* __restrict__ W2,
                     float* __restrict__ out,
                     int totalRows)
{
    const int tid  = threadIdx.x;
    const int wv   = tid >> 5;
    const int lane = tid & 31;

    // first global row handled by this wave (multiple of 16 -> sample aligned)
    const int g0 = (blockIdx.x * WAVES + wv) * ROWS_PER_WAVE;

    float* zt  = lds + wv * PW_FLOATS;
    float* sc  = zt + ZT_FLOATS;     // 16 scores
    float* bet = sc + 16;            // 16 x (b1,b2)

    // ---- async DMA: 16 rows x 256 f32 of z -> LDS (b128 per lane, ASYNCcnt) ----
    {
        const unsigned ldsBase = (unsigned)(wv * PW_FLOATS) * 4u; // dyn-LDS offset of zt
        const float* src = z + (size_t)g0 * DIM;
        #pragma unroll
        for (int i = lane; i < ROWS_PER_WAVE * (DIM / 4); i += 32) {
            const int row  = i >> 6;          // 64 float4 per row
            const int col4 = i & 63;
            if (g0 + row < totalRows) {
                const unsigned ldst = ldsBase + (unsigned)(row * STRIDE + col4 * 4) * 4u;
                const float*   ga   = src + i * 4;
                asm volatile("global_load_async_to_lds_b128 %0, %1, off"
                             :: "v"(ldst), "v"(ga) : "memory");
            }
        }
    }
    wave_fence_async();

    // ---- hidden[16x16] = zt(16x256) @ W1^T(256x16) via 64x WMMA f32 16x16x4 ----
    // A: lane l holds row m=l&15, K pair kb..kb+1, kb=(l>>4)*2 (LDS, conflict-free)
    // B: lane l holds col h=l&15,  K pair kb..kb+1 (W1 global, WGP$/L2-resident)
    const int m  = lane & 15;
    const int kb = (lane >> 4) << 1;

    const float* aptr = zt + m * STRIDE + kb;
    const float* bptr = W1 + m * DIM + kb;     // h == m == lane&15 for B

    v8f acc = {0.f, 0.f, 0.f, 0.f, 0.f, 0.f, 0.f, 0.f};
    #pragma unroll 8
    for (int d0 = 0; d0 < DIM; d0 += 4) {
        v2f a = *(const v2f*)(aptr + d0);
        v2f b = *(const v2f*)(bptr + d0);
        acc = __builtin_amdgcn_wmma_f32_16x16x4_f32(false, a, false, b,
                                                    (short)0, acc, false, false);
    }

    // ---- scores: s[row] = W2 . leaky_relu(hidden[row]) ----
    // C layout: VGPR v -> M = v (lanes 0-15) / v+8 (lanes 16-31); N = lane&15.
    const float w2v = W2[m];
    #pragma unroll
    for (int v = 0; v < 8; ++v) {
        float x = acc[v];
        float h = (x > 0.0f) ? x : 0.01f * x;       // leaky_relu, slope .01
        float t = half16_reduce_add(h * w2v);
        if (m == 0)
            sc[v + ((lane >> 4) << 3)] = t;         // lane0 -> row v, lane16 -> row v+8
    }
    wave_fence_ds();

    // ---- vectorized softmax: lane m (0..15) handles row m, betas -> LDS ----
    if (lane < 16) {
        const int r = lane & 7;
        const int b = lane & 8;
        const int k1 = (r == 0) ? 1 : ((r == RNUM - 1) ? RNUM - 2 : r + 1);
        const int k2 = (r == 0) ? 2 : ((r == RNUM - 1) ? RNUM - 3 : r - 1);
        const float s0 = sc[b + r];
        const float s1 = sc[b + k1];
        const float s2 = sc[b + k2];
        const float mx = fmaxf(s0, fmaxf(s1, s2));
        const float e0 = __expf(s0 - mx);
        const float e1 = __expf(s1 - mx);
        const float e2 = __expf(s2 - mx);
        const float inv = 1.0f / (e0 + e1 + e2);
        float2 bb; bb.x = e1 * inv; bb.y = e2 * inv;
        *(float2*)(bet + 2 * lane) = bb;
    }
    wave_fence_ds();

    // ---- per row: weighted neighbor sum + residual (coalesced b128 stores) ----
    #pragma unroll
    for (int m2 = 0; m2 < ROWS_PER_WAVE; ++m2) {
        if (g0 + m2 >= totalRows) continue;     // wave-uniform guard (tail block)
        const int r    = m2 & 7;
        const int base = m2 & 8;
        const int k1 = (r == 0) ? 1 : ((r == RNUM - 1) ? RNUM - 2 : r + 1);
        const int k2 = (r == 0) ? 2 : ((r == RNUM - 1) ? RNUM - 3 : r - 1);

        const float2 bb = *(const float2*)(bet + 2 * m2);  // LDS broadcast
        const float b1 = bb.x, b2 = bb.y;

        const float* zr = zt + (base + r ) * STRIDE;
        const float* z1 = zt + (base + k1) * STRIDE;
        const float* z2 = zt + (base + k2) * STRIDE;
        float* op = out + (size_t)(g0 + m2) * DIM;

        #pragma unroll
        for (int halfc = 0; halfc < 2; ++halfc) {
            const int c = lane * 4 + halfc * 128;   // 512B contiguous per store phase
            float4 a = *(const float4*)(zr + c);
            float4 p = *(const float4*)(z1 + c);
            float4 q = *(const float4*)(z2 + c);
            float4 o;
            o.x = a.x + b1 * p.x + b2 * q.x;
            o.y = a.y + b1 * p.y + b2 * q.y;
            o.z = a.z + b1 * p.z + b2 * q.z;
            o.w = a.w + b1 * p.w + b2 * q.w;
            *(float4*)(op + c) = o;
        }
    }
}

extern "C" void kernel_launch(void* const* d_in, const int* in_sizes, int n_in,
                              void* d_out, int out_size, void* d_ws, size_t ws_size,
                              hipStream_t stream) {
    const float* z  = (const float*)d_in[0];
    const float* W1 = (const float*)d_in[1];
    const float* W2 = (const float*)d_in[2];
    float* out = (float*)d_out;

    const int totalRows    = in_sizes[0] / DIM;            // N * R = 400000
    const int rowsPerBlock = WAVES * ROWS_PER_WAVE;        // 128
    const int grid         = (totalRows + rowsPerBlock - 1) / rowsPerBlock;

    const size_t ldsBytes = (size_t)(WAVES * PW_FLOATS) * sizeof(float); // ~131.5 KB

    // >64KB dynamic LDS: legal on CDNA5 (320KB/WGP); opt in explicitly.
    hipFuncSetAttribute(reinterpret_cast<const void*>(ring_attn_fused),
                        hipFuncAttributeMaxDynamicSharedMemorySize,
                        (int)ldsBytes);

    ring_attn_fused<<<grid, WAVES * 32, ldsBytes, stream>>>(z, W1, W2, out, totalRows);
}